// BatchedModelManifoldGeodesicFlow_62251255988473
// MI455X (gfx1250) — compile-verified
//
#include <hip/hip_runtime.h>
#include <hip/hip_bf16.h>
#include <math.h>

typedef float v2f __attribute__((ext_vector_type(2)));
typedef float v8f __attribute__((ext_vector_type(8)));

#define DD 4096
#define HH 512
#define NC 64
#define CC 64
#define NB 8
#define COUT 8

// workspace offsets in floats (all 64-float aligned)
#define OFF_CHUNKA 0         // 64*512
#define OFF_W1S    32768     // 512
#define OFF_M      33280     // 512*512
#define OFF_X      295424    // 8*4096
#define OFF_VNORM  328192    // 8 (padded)
#define OFF_DV     328256    // 8*512
#define OFF_EV     332352    // 8*512
#define OFF_RV     336448    // 8*512
#define OFF_AD     340544    // 8*64*512
#define OFF_G      602688    // 8*64*512
#define OFF_GBUF   864832    // 8*64*4096
#define OFF_UPART  2961984   // 8*8*512  -> total 2994752 floats (~12 MB)

// ---- A[n,h] = sum of 64 rows of W1 in chunk n -----------------------------
__global__ void mmgf_chunkA(const float* __restrict__ W1, float* __restrict__ ws) {
    int idx = blockIdx.x * 256 + threadIdx.x;      // 32768 = 64*512
    int n = idx >> 9, h = idx & 511;
    const float* p = W1 + (size_t)(n * 64) * HH + h;
    float s = 0.f;
    #pragma unroll 4
    for (int j = 0; j < 64; ++j) s += p[(size_t)j * HH];
    ws[OFF_CHUNKA + idx] = s;
}

// ---- w1s[h] = sum_n A[n,h] ------------------------------------------------
__global__ void mmgf_w1s(float* __restrict__ ws) {
    int h = blockIdx.x * 256 + threadIdx.x;        // 512
    float s = 0.f;
    for (int n = 0; n < NC; ++n) s += ws[OFF_CHUNKA + n * HH + h];
    ws[OFF_W1S + h] = s;
}

// ---- M = W2 @ W2^T (symmetric 512x512) ------------------------------------
__global__ void mmgf_M(const float* __restrict__ W2, float* __restrict__ ws) {
    int idx = blockIdx.x * 256 + threadIdx.x;      // 262144
    int h = idx >> 9, hp = idx & 511;
    float s = 0.f;
    #pragma unroll
    for (int c = 0; c < COUT; ++c) s += W2[h * COUT + c] * W2[hp * COUT + c];
    ws[OFF_M + idx] = s;
}

// ---- x, ||v||, copy velocity to first half of output ----------------------
__global__ void mmgf_x(const float* __restrict__ tptr, const float* __restrict__ state,
                       const float* __restrict__ x0, const float* __restrict__ x1,
                       float* __restrict__ ws, float* __restrict__ out) {
    int s = blockIdx.x, tid = threadIdx.x;
    float ts = tptr[0];
    float window = 4.f * ts * (1.f - ts);
    float acc = 0.f;
    for (int j = tid; j < DD; j += 256) {
        float dev = state[s * DD + j];
        float v   = state[(NB + s) * DD + j];
        float xv  = x0[s * DD + j] + ts * (x1[s * DD + j] - x0[s * DD + j]) + window * dev;
        ws[OFF_X + s * DD + j] = xv;
        out[s * DD + j] = v;                       // first half of output
        acc = fmaf(v, v, acc);
    }
    __shared__ float red[256];
    red[tid] = acc; __syncthreads();
    for (int off = 128; off > 0; off >>= 1) {
        if (tid < off) red[tid] += red[tid + off];
        __syncthreads();
    }
    if (tid == 0) ws[OFF_VNORM + s] = sqrtf(red[0]);
}

// ---- partial u: upart[s][q][h] = sum_{j in slice q} x[s,j]*W1[j,h] --------
__global__ void mmgf_upart(const float* __restrict__ W1, float* __restrict__ ws) {
    int blk = blockIdx.x;                  // 64 = 8 samples * 8 slices
    int s = blk >> 3, q = blk & 7;
    int h = threadIdx.x;                   // 512
    __shared__ float xsh[512];
    xsh[h] = ws[OFF_X + s * DD + q * 512 + h];
    __syncthreads();
    float u = 0.f;
    const float* Wp = W1 + (size_t)(q * 512) * HH + h;
    #pragma unroll 8
    for (int j = 0; j < 512; ++j) u = fmaf(xsh[j], Wp[(size_t)j * HH], u);
    ws[OFF_UPART + blk * HH + h] = u;
}

// ---- combine partials, activation terms d,e,r and Ad = A * d --------------
__global__ void mmgf_act(const float* __restrict__ W2, float* __restrict__ ws) {
    int s = blockIdx.x, h = threadIdx.x;           // 512 threads = one per h
    __shared__ float sacc[COUT];
    if (h < COUT) sacc[h] = 0.f;
    float u = 0.f;
    #pragma unroll
    for (int q = 0; q < 8; ++q) u += ws[OFF_UPART + (s * 8 + q) * HH + h];
    __syncthreads();
    float hh = tanhf(u);
    float dd = 1.f - hh * hh;
    float w1sh = ws[OFF_W1S + h];
    float t1 = w1sh * dd;
    #pragma unroll
    for (int c = 0; c < COUT; ++c) atomicAdd(&sacc[c], t1 * W2[h * COUT + c]);
    __syncthreads();
    float q2 = 0.f;
    #pragma unroll
    for (int c = 0; c < COUT; ++c) q2 = fmaf(W2[h * COUT + c], sacc[c], q2);
    float m2hd = -2.f * hh * dd;
    ws[OFF_DV + s * HH + h] = dd;
    ws[OFF_EV + s * HH + h] = m2hd * q2;
    ws[OFF_RV + s * HH + h] = m2hd * w1sh;
    float* Ad = ws + OFF_AD + s * NC * HH;
    for (int n = 0; n < NC; ++n)
        Ad[n * HH + h] = ws[OFF_CHUNKA + n * HH + h] * dd;
}

// ---- WMMA (2x2 blocked): C0 = Ad @ M ; G = A*e + C0*r ---------------------
// 256 waves: 8 samples x (2 row-groups x 16 col-groups), 32x32 out per wave.
__global__ void mmgf_G(float* __restrict__ ws) {
    int wv   = blockIdx.x * 8 + (threadIdx.x >> 5);   // 0..255
    int lane = threadIdx.x & 31;
    int s    = wv >> 5;
    int t    = wv & 31;
    int rg   = (t >> 4) << 5;        // row base: 0 or 32
    int cg   = (t & 15) << 5;        // col base: 0..480 step 32
    int m    = lane & 15;
    int kb   = (lane >> 4) << 1;
    const float* Ad = ws + OFF_AD + s * NC * HH;
    const float* Mm = ws + OFF_M;
    v8f a00 = {0.f,0.f,0.f,0.f,0.f,0.f,0.f,0.f};
    v8f a01 = a00, a10 = a00, a11 = a00;
    #pragma unroll 2
    for (int kk = 0; kk < HH / 4; ++kk) {
        int h0 = kk * 4 + kb;
        v2f fa0 = *(const v2f*)(Ad + (rg + m) * HH + h0);
        v2f fa1 = *(const v2f*)(Ad + (rg + 16 + m) * HH + h0);
        v2f fb0 = *(const v2f*)(Mm + (cg + m) * HH + h0);       // M symmetric
        v2f fb1 = *(const v2f*)(Mm + (cg + 16 + m) * HH + h0);
        a00 = __builtin_amdgcn_wmma_f32_16x16x4_f32(false, fa0, false, fb0, (short)0, a00, false, false);
        a01 = __builtin_amdgcn_wmma_f32_16x16x4_f32(false, fa0, false, fb1, (short)0, a01, false, false);
        a10 = __builtin_amdgcn_wmma_f32_16x16x4_f32(false, fa1, false, fb0, (short)0, a10, false, false);
        a11 = __builtin_amdgcn_wmma_f32_16x16x4_f32(false, fa1, false, fb1, (short)0, a11, false, false);
    }
    const float* ev = ws + OFF_EV + s * HH;
    const float* rv = ws + OFF_RV + s * HH;
    const float* cA = ws + OFF_CHUNKA;
    float* G = ws + OFF_G + s * NC * HH;
    int col0 = cg + (lane & 15), col1 = col0 + 16;
    int rb   = rg + ((lane >> 4) << 3);
    float e0 = ev[col0], r0 = rv[col0], e1 = ev[col1], r1 = rv[col1];
    #pragma unroll
    for (int rr = 0; rr < 8; ++rr) {
        int row0 = rb + rr, row1 = row0 + 16;
        G[row0 * HH + col0] = cA[row0 * HH + col0] * e0 + a00[rr] * r0;
        G[row0 * HH + col1] = cA[row0 * HH + col1] * e1 + a01[rr] * r1;
        G[row1 * HH + col0] = cA[row1 * HH + col0] * e0 + a10[rr] * r0;
        G[row1 * HH + col1] = cA[row1 * HH + col1] * e1 + a11[rr] * r1;
    }
}

// ---- WMMA (2x2 blocked): g = G @ W1^T  (64x4096 per sample) ---------------
// 2048 waves: 8 samples x (2 row-groups x 128 col-groups), 32x32 per wave.
__global__ void mmgf_gemm(const float* __restrict__ W1, float* __restrict__ ws) {
    int wv   = blockIdx.x * 8 + (threadIdx.x >> 5);   // 0..2047
    int lane = threadIdx.x & 31;
    int s    = wv >> 8;
    int t    = wv & 255;
    int rg   = (t >> 7) << 5;        // 0 or 32
    int cg   = (t & 127) << 5;       // 0..4064 step 32
    int m    = lane & 15;
    int kb   = (lane >> 4) << 1;
    const float* G = ws + OFF_G + s * NC * HH;
    v8f a00 = {0.f,0.f,0.f,0.f,0.f,0.f,0.f,0.f};
    v8f a01 = a00, a10 = a00, a11 = a00;
    #pragma unroll 2
    for (int kk = 0; kk < HH / 4; ++kk) {
        int h0 = kk * 4 + kb;
        v2f fa0 = *(const v2f*)(G + (rg + m) * HH + h0);
        v2f fa1 = *(const v2f*)(G + (rg + 16 + m) * HH + h0);
        v2f fb0 = *(const v2f*)(W1 + (size_t)(cg + m) * HH + h0);       // W1^T frag = W1 row
        v2f fb1 = *(const v2f*)(W1 + (size_t)(cg + 16 + m) * HH + h0);
        a00 = __builtin_amdgcn_wmma_f32_16x16x4_f32(false, fa0, false, fb0, (short)0, a00, false, false);
        a01 = __builtin_amdgcn_wmma_f32_16x16x4_f32(false, fa0, false, fb1, (short)0, a01, false, false);
        a10 = __builtin_amdgcn_wmma_f32_16x16x4_f32(false, fa1, false, fb0, (short)0, a10, false, false);
        a11 = __builtin_amdgcn_wmma_f32_16x16x4_f32(false, fa1, false, fb1, (short)0, a11, false, false);
    }
    float* g = ws + OFF_GBUF + s * NC * DD;
    int col0 = cg + (lane & 15), col1 = col0 + 16;
    int rb   = rg + ((lane >> 4) << 3);
    #pragma unroll
    for (int rr = 0; rr < 8; ++rr) {
        int row0 = rb + rr, row1 = row0 + 16;
        g[row0 * DD + col0] = a00[rr];
        g[row0 * DD + col1] = a01[rr];
        g[row1 * DD + col0] = a10[rr];
        g[row1 * DD + col1] = a11[rr];
    }
}

// ---- normalize rows, symmetrize, contract with v, finalize ----------------
__global__ void mmgf_finish(const float* __restrict__ state, float* __restrict__ ws,
                            float* __restrict__ out) {
    int blk = blockIdx.x;            // 512 = 8 samples * 64 chunks
    int s = blk >> 6, n = blk & 63;
    int i = threadIdx.x;             // 64 threads
    __shared__ float gn[64 * 65];    // padded to dodge bank conflicts
    __shared__ float vsh[64];
    const float* g = ws + OFF_GBUF + s * NC * DD + n * DD + i * CC;
    float ss = 0.f;
    for (int k = 0; k < CC; ++k) { float x = g[k]; ss = fmaf(x, x, ss); }
    float inv = 1.f / (sqrtf(ss) + 1e-6f);
    for (int k = 0; k < CC; ++k) gn[i * 65 + k] = g[k] * inv;
    vsh[i] = state[(NB + s) * DD + n * CC + i];
    __syncthreads();
    float a1 = 0.f, a2 = 0.f;
    for (int k = 0; k < CC; ++k) {
        float vk = vsh[k];
        a1 = fmaf(gn[i * 65 + k], vk, a1);
        a2 = fmaf(gn[k * 65 + i], vk, a2);
    }
    float a = -0.5f * vsh[i] * (a1 + a2);
    float vn = ws[OFF_VNORM + s];
    float dev = state[s * DD + n * CC + i];
    out[(NB + s) * DD + n * CC + i] = a / (vn + 1e-6f) - 0.1f * dev;
}

extern "C" void kernel_launch(void* const* d_in, const int* in_sizes, int n_in,
                              void* d_out, int out_size, void* d_ws, size_t ws_size,
                              hipStream_t stream) {
    const float* t     = (const float*)d_in[0];
    const float* state = (const float*)d_in[1];
    const float* x0    = (const float*)d_in[2];
    const float* x1    = (const float*)d_in[3];
    const float* W1    = (const float*)d_in[4];
    const float* W2    = (const float*)d_in[5];
    float* out = (float*)d_out;
    float* ws  = (float*)d_ws;

    mmgf_chunkA<<<128, 256, 0, stream>>>(W1, ws);
    mmgf_w1s   <<<2,   256, 0, stream>>>(ws);
    mmgf_M     <<<1024,256, 0, stream>>>(W2, ws);
    mmgf_x     <<<8,   256, 0, stream>>>(t, state, x0, x1, ws, out);
    mmgf_upart <<<64,  512, 0, stream>>>(W1, ws);
    mmgf_act   <<<8,   512, 0, stream>>>(W2, ws);
    mmgf_G     <<<32,  256, 0, stream>>>(ws);
    mmgf_gemm  <<<256, 256, 0, stream>>>(W1, ws);
    mmgf_finish<<<512, 64,  0, stream>>>(state, ws, out);
}